// GAT_32796370272386
// MI455X (gfx1250) — compile-verified
//
#include <hip/hip_runtime.h>
#include <hip/hip_bf16.h>
#include <math.h>

typedef __attribute__((ext_vector_type(2))) float v2f;
typedef __attribute__((ext_vector_type(8))) float v8f;

#define TPB 256
#define MAXK 128
#define A_PITCH (MAXK + 4)   // pad: row stride 132 dwords -> banks 4 apart

// ---------------------------------------------------------------------------
// Float atomic max via sign-aware integer atomics (monotone bit trick).
// ---------------------------------------------------------------------------
__device__ __forceinline__ void atomicMaxF32(float* addr, float val) {
  if (val >= 0.0f) {
    atomicMax((int*)addr, __float_as_int(val));
  } else {
    atomicMin((unsigned int*)addr, __float_as_uint(val));
  }
}

// ---------------------------------------------------------------------------
// Fill
// ---------------------------------------------------------------------------
__global__ void __launch_bounds__(TPB) fill_f32(float* __restrict__ p, float v,
                                                long long n) {
  long long i = (long long)blockIdx.x * blockDim.x + threadIdx.x;
  if (i < n) p[i] = v;
}

// ---------------------------------------------------------------------------
// C = X @ W + bias  via V_WMMA_F32_16X16X4_F32, A-tile staged in LDS.
// X: [M,K] row-major, W: [K,Nc] row-major, out: [M,Nc].
// Block: 8 waves; one tm (16 rows) per block, wave w -> tile tn0+w.
// A layout (ISA 7.12.2, 32-bit A 16x4): lane&15 = row, lane>>4 selects
// K-pair {2h, 2h+1} in VGPR0/1. B mirrors with columns.
// ---------------------------------------------------------------------------
__global__ void __launch_bounds__(TPB) gemm_bias_wmma(
    const float* __restrict__ X, const float* __restrict__ W,
    const float* __restrict__ bias, float* __restrict__ out,
    int M, int K, int Nc) {
  __shared__ float As[16 * A_PITCH];

  const int lane   = threadIdx.x & 31;
  const int wave   = threadIdx.x >> 5;
  const int tilesN = Nc >> 4;
  const int tnb    = (tilesN + 7) >> 3;           // tn-groups per tm
  const int tm     = blockIdx.x / tnb;
  const int tn     = (blockIdx.x % tnb) * 8 + wave;
  const int half   = lane >> 4;
  const int l15    = lane & 15;
  const int rowA0  = tm * 16;

  // ---- Stage A-tile (16 x K) into LDS ----
#if defined(__gfx1250__)
  if (K == MAXK) {
    // Each wave async-copies 2 rows; each lane moves 16B.
    // Generic LDS pointers carry the LDS byte offset in their low 32 bits.
#pragma unroll
    for (int r = 0; r < 2; ++r) {
      const int row = wave * 2 + r;
      const unsigned long long gaddr = (unsigned long long)(uintptr_t)(
          X + (size_t)(rowA0 + row) * K + lane * 4);
      const unsigned laddr = (unsigned)(uintptr_t)(const void*)
          &As[row * A_PITCH + lane * 4];
      asm volatile("global_load_async_to_lds_b128 %0, %1, off"
                   :: "v"(laddr), "v"(gaddr) : "memory");
    }
    asm volatile("s_wait_asynccnt 0x0" ::: "memory");
  } else
#endif
  {
    const int nvec = 16 * (K >> 2);               // float4 count
    for (int idx = threadIdx.x; idx < nvec; idx += TPB) {
      const int row = idx / (K >> 2);
      const int cv  = idx % (K >> 2);
      const float4 v =
          *(const float4*)(X + (size_t)(rowA0 + row) * K + cv * 4);
      *(float4*)(&As[row * A_PITCH + cv * 4]) = v;
    }
  }
  __syncthreads();

  if (tn >= tilesN) return;                       // wave-uniform

  const float* __restrict__ arow = &As[l15 * A_PITCH];
  const int colB = tn * 16 + l15;

  v8f acc = {};
  for (int k0 = 0; k0 < K; k0 += 4) {
    const int ka = k0 + half * 2;
    v2f a, b;
    a.x = arow[ka];                               // ds_load, conflict-free
    a.y = arow[ka + 1];
    b.x = W[(size_t)ka * Nc + colB];
    b.y = W[(size_t)(ka + 1) * Nc + colB];
    acc = __builtin_amdgcn_wmma_f32_16x16x4_f32(
        /*neg_a=*/false, a, /*neg_b=*/false, b,
        /*c_mod=*/(short)0, acc, /*reuse_a=*/false, /*reuse_b=*/false);
  }

  const float bv = bias[colB];
  // C/D layout: VGPR v holds M = v + 8*half, N = lane&15.
#pragma unroll
  for (int v = 0; v < 8; ++v) {
    const int row = rowA0 + v + half * 8;
    out[(size_t)row * Nc + colB] = acc[v] + bv;
  }
}

// ---------------------------------------------------------------------------
// Edge helpers: edges 0..E-1 from edge_index, edges E..E+Nn-1 are self loops.
// ---------------------------------------------------------------------------
__device__ __forceinline__ void edge_endpoints(const int* __restrict__ ei,
                                               long long e, int E,
                                               int& src, int& dst) {
  if (e < E) {
    src = ei[2 * e];
    dst = ei[2 * e + 1];
  } else {
    src = dst = (int)(e - E);
  }
}

// alpha[e,h] = att[h,:] . leaky_relu(xr[dst,h,:] + xl[src,h,:], 0.2)
// + running segment max into m[dst,h].  (float4 vectorized gathers)
__global__ void __launch_bounds__(TPB) edge_scores(
    const float* __restrict__ xl, const float* __restrict__ xr,
    const float* __restrict__ att, const int* __restrict__ ei,
    int E, int Nn, int H, int C,
    float* __restrict__ alpha, float* __restrict__ m) {
  long long g = (long long)blockIdx.x * blockDim.x + threadIdx.x;
  long long total = (long long)(E + Nn) * H;
  if (g >= total) return;
  const int h = (int)(g % H);
  const long long e = g / H;
  int src, dst;
  edge_endpoints(ei, e, E, src, dst);
  const float4* __restrict__ pl =
      (const float4*)(xl + ((size_t)src * H + h) * C);
  const float4* __restrict__ pr =
      (const float4*)(xr + ((size_t)dst * H + h) * C);
  const float4* __restrict__ pa = (const float4*)(att + (size_t)h * C);
  float sum = 0.0f;
  const int C4 = C >> 2;
  for (int c = 0; c < C4; ++c) {
    const float4 l = pl[c];
    const float4 r = pr[c];
    const float4 a = pa[c];
    float v0 = r.x + l.x, v1 = r.y + l.y, v2 = r.z + l.z, v3 = r.w + l.w;
    v0 = (v0 > 0.0f) ? v0 : 0.2f * v0;
    v1 = (v1 > 0.0f) ? v1 : 0.2f * v1;
    v2 = (v2 > 0.0f) ? v2 : 0.2f * v2;
    v3 = (v3 > 0.0f) ? v3 : 0.2f * v3;
    sum += a.x * v0 + a.y * v1 + a.z * v2 + a.w * v3;
  }
  alpha[e * H + h] = sum;
  atomicMaxF32(m + (size_t)dst * H + h, sum);
}

// alpha[e,h] <- exp(alpha[e,h] - m[dst,h]); s[dst,h] += alpha[e,h]
__global__ void __launch_bounds__(TPB) edge_exp(
    float* __restrict__ alpha, const float* __restrict__ m,
    float* __restrict__ s, const int* __restrict__ ei,
    int E, int Nn, int H) {
  long long g = (long long)blockIdx.x * blockDim.x + threadIdx.x;
  long long total = (long long)(E + Nn) * H;
  if (g >= total) return;
  const int h = (int)(g % H);
  const long long e = g / H;
  int src, dst;
  edge_endpoints(ei, e, E, src, dst);
  float mv = m[(size_t)dst * H + h];
  if (!__builtin_isfinite(mv)) mv = 0.0f;
  const float a = __expf(alpha[e * H + h] - mv);
  alpha[e * H + h] = a;
  atomicAdd(s + (size_t)dst * H + h, a);
}

// acc[dst, hc] += (alpha[e,h] / (s[dst,h]+1e-16)) * xl[src, hc]
__global__ void __launch_bounds__(TPB) edge_aggregate(
    const float* __restrict__ alpha, const float* __restrict__ s,
    const float* __restrict__ xl, const int* __restrict__ ei,
    int E, int Nn, int H, int C, float* __restrict__ acc) {
  long long g = (long long)blockIdx.x * blockDim.x + threadIdx.x;
  const int HC = H * C;
  long long total = (long long)(E + Nn) * HC;
  if (g >= total) return;
  const int hc = (int)(g % HC);
  const long long e = g / HC;
  const int h = hc / C;
  int src, dst;
  edge_endpoints(ei, e, E, src, dst);
  const float w = alpha[e * H + h] /
                  (s[(size_t)dst * H + h] + 1e-16f);
  atomicAdd(acc + (size_t)dst * HC + hc, w * xl[(size_t)src * HC + hc]);
}

// h = elu(acc + bias) in place
__global__ void __launch_bounds__(TPB) elu_bias(float* __restrict__ acc,
                                                const float* __restrict__ bias,
                                                long long n, int F) {
  long long i = (long long)blockIdx.x * blockDim.x + threadIdx.x;
  if (i >= n) return;
  float v = acc[i] + bias[i % F];
  acc[i] = (v > 0.0f) ? v : (expf(v) - 1.0f);
}

// out = acc + bias
__global__ void __launch_bounds__(TPB) add_bias_out(
    const float* __restrict__ acc, const float* __restrict__ bias,
    float* __restrict__ out, long long n, int F) {
  long long i = (long long)blockIdx.x * blockDim.x + threadIdx.x;
  if (i >= n) return;
  out[i] = acc[i] + bias[i % F];
}

// ---------------------------------------------------------------------------
// Host-side orchestration
// ---------------------------------------------------------------------------
static inline unsigned nblk(long long n) {
  return (unsigned)((n + TPB - 1) / TPB);
}

extern "C" void kernel_launch(void* const* d_in, const int* in_sizes, int n_in,
                              void* d_out, int out_size, void* d_ws,
                              size_t ws_size, hipStream_t stream) {
  const int DIN = 128, DH = 32, HEADS = 4, DOUT = 64;
  const int F1 = HEADS * DH;  // 128

  const float* x    = (const float*)d_in[0];
  const int*   ei   = (const int*)d_in[1];
  const float* W1l  = (const float*)d_in[2];
  const float* b1l  = (const float*)d_in[3];
  const float* W1r  = (const float*)d_in[4];
  const float* b1r  = (const float*)d_in[5];
  const float* att1 = (const float*)d_in[6];
  const float* bias1= (const float*)d_in[7];
  const float* W2l  = (const float*)d_in[8];
  const float* b2l  = (const float*)d_in[9];
  const float* W2r  = (const float*)d_in[10];
  const float* b2r  = (const float*)d_in[11];
  const float* att2 = (const float*)d_in[12];
  const float* bias2= (const float*)d_in[13];
  float* out = (float*)d_out;

  const int N  = in_sizes[0] / DIN;   // 50000
  const int E  = in_sizes[1] / 2;     // 800000
  const int ET = E + N;               // with self loops

  // Workspace layout (floats)
  float* ws = (float*)d_ws;
  size_t o = 0;
  float* xl1    = ws + o; o += (size_t)N * F1;
  float* xr1    = ws + o; o += (size_t)N * F1;
  float* acc1   = ws + o; o += (size_t)N * F1;   // becomes h after elu
  float* xl2    = ws + o; o += (size_t)N * DOUT;
  float* xr2    = ws + o; o += (size_t)N * DOUT;
  float* acc2   = ws + o; o += (size_t)N * DOUT;
  float* alpha1 = ws + o; o += (size_t)ET * HEADS;
  float* alpha2 = ws + o; o += (size_t)ET;
  float* m1     = ws + o; o += (size_t)N * HEADS;
  float* s1     = ws + o; o += (size_t)N * HEADS;
  float* m2     = ws + o; o += (size_t)N;
  float* s2     = ws + o; o += (size_t)N;
  (void)ws_size; (void)n_in; (void)out_size;

  const float NEG_INF = -__builtin_huge_valf();

  // Init reductions / accumulators
  fill_f32<<<nblk((long long)N * HEADS), TPB, 0, stream>>>(m1, NEG_INF, (long long)N * HEADS);
  fill_f32<<<nblk((long long)N * HEADS), TPB, 0, stream>>>(s1, 0.0f, (long long)N * HEADS);
  fill_f32<<<nblk((long long)N * F1),    TPB, 0, stream>>>(acc1, 0.0f, (long long)N * F1);
  fill_f32<<<nblk(N), TPB, 0, stream>>>(m2, NEG_INF, (long long)N);
  fill_f32<<<nblk(N), TPB, 0, stream>>>(s2, 0.0f, (long long)N);
  fill_f32<<<nblk((long long)N * DOUT),  TPB, 0, stream>>>(acc2, 0.0f, (long long)N * DOUT);

  // ---- Layer 1 ----
  {
    const int tilesM = N / 16, tilesN = F1 / 16;
    const unsigned blocks = (unsigned)(tilesM * ((tilesN + 7) / 8));
    gemm_bias_wmma<<<blocks, TPB, 0, stream>>>(x, W1l, b1l, xl1, N, DIN, F1);
    gemm_bias_wmma<<<blocks, TPB, 0, stream>>>(x, W1r, b1r, xr1, N, DIN, F1);
  }
  edge_scores<<<nblk((long long)ET * HEADS), TPB, 0, stream>>>(
      xl1, xr1, att1, ei, E, N, HEADS, DH, alpha1, m1);
  edge_exp<<<nblk((long long)ET * HEADS), TPB, 0, stream>>>(
      alpha1, m1, s1, ei, E, N, HEADS);
  edge_aggregate<<<nblk((long long)ET * F1), TPB, 0, stream>>>(
      alpha1, s1, xl1, ei, E, N, HEADS, DH, acc1);
  elu_bias<<<nblk((long long)N * F1), TPB, 0, stream>>>(
      acc1, bias1, (long long)N * F1, F1);

  // ---- Layer 2 ----
  {
    const int tilesM = N / 16, tilesN = DOUT / 16;
    const unsigned blocks = (unsigned)(tilesM * ((tilesN + 7) / 8));
    gemm_bias_wmma<<<blocks, TPB, 0, stream>>>(acc1, W2l, b2l, xl2, N, F1, DOUT);
    gemm_bias_wmma<<<blocks, TPB, 0, stream>>>(acc1, W2r, b2r, xr2, N, F1, DOUT);
  }
  edge_scores<<<nblk((long long)ET), TPB, 0, stream>>>(
      xl2, xr2, att2, ei, E, N, 1, DOUT, alpha2, m2);
  edge_exp<<<nblk((long long)ET), TPB, 0, stream>>>(
      alpha2, m2, s2, ei, E, N, 1);
  edge_aggregate<<<nblk((long long)ET * DOUT), TPB, 0, stream>>>(
      alpha2, s2, xl2, ei, E, N, 1, DOUT, acc2);
  add_bias_out<<<nblk((long long)N * DOUT), TPB, 0, stream>>>(
      acc2, bias2, out, (long long)N * DOUT, DOUT);
}